// SetAbstractionMSG_8478265442717
// MI455X (gfx1250) — compile-verified
//
#include <hip/hip_runtime.h>

// Problem constants from the reference
#define BB 8
#define NN 4096
#define PP 1024
#define CIN 64
#define KMAX 64
#define EPS_BN 1e-5f

typedef __attribute__((ext_vector_type(2))) float v2f;
typedef __attribute__((ext_vector_type(8))) float v8f;

// ---------------------------------------------------------------------------
// 1) Gather sampled centers -> new_xyz output (B,3,P)
// ---------------------------------------------------------------------------
__global__ void k_newxyz(const float* __restrict__ xyz, const int* __restrict__ fps,
                         float* __restrict__ out) {
    int i = blockIdx.x * blockDim.x + threadIdx.x;
    if (i >= BB * 3 * PP) return;
    int p = i % PP;
    int c = (i / PP) % 3;
    int b = i / (3 * PP);
    out[i] = xyz[(b * 3 + c) * NN + fps[p]];
}

// ---------------------------------------------------------------------------
// 2) kNN: one workgroup per (b,p); distances in LDS, extract 64 sorted minima
// ---------------------------------------------------------------------------
__global__ void __launch_bounds__(256) k_knn(const float* __restrict__ xyz,
                                             const int* __restrict__ fps,
                                             int* __restrict__ knn) {
    int bp = blockIdx.x;
    int b = bp / PP;
    int p = bp % PP;
    int q = fps[p];
    float qx = xyz[(b * 3 + 0) * NN + q];
    float qy = xyz[(b * 3 + 1) * NN + q];
    float qz = xyz[(b * 3 + 2) * NN + q];

    __shared__ float dsh[NN];
    __shared__ float bv[256];
    __shared__ int   bi[256];

    for (int n = threadIdx.x; n < NN; n += 256) {
        float dx = xyz[(b * 3 + 0) * NN + n] - qx;
        float dy = xyz[(b * 3 + 1) * NN + n] - qy;
        float dz = xyz[(b * 3 + 2) * NN + n] - qz;
        dsh[n] = dx * dx + dy * dy + dz * dz;
    }
    __syncthreads();

    for (int r = 0; r < KMAX; ++r) {
        float mv = 3.4e38f;
        int   mi = NN;
        for (int n = threadIdx.x; n < NN; n += 256) {
            float v = dsh[n];
            if (v < mv || (v == mv && n < mi)) { mv = v; mi = n; }
        }
        bv[threadIdx.x] = mv;
        bi[threadIdx.x] = mi;
        __syncthreads();
        for (int s = 128; s > 0; s >>= 1) {
            if (threadIdx.x < s) {
                float ov = bv[threadIdx.x + s];
                int   oi = bi[threadIdx.x + s];
                if (ov < bv[threadIdx.x] ||
                    (ov == bv[threadIdx.x] && oi < bi[threadIdx.x])) {
                    bv[threadIdx.x] = ov;
                    bi[threadIdx.x] = oi;
                }
            }
            __syncthreads();
        }
        if (threadIdx.x == 0) {
            knn[bp * KMAX + r] = bi[0];
            dsh[bi[0]] = 3.4e38f;
        }
        __syncthreads();
    }
}

// ---------------------------------------------------------------------------
// 3) Build H0: (M x 68) rows m=((b*P+p)*K+k): [dxyz(3) | feats(64) | pad(1)]
// ---------------------------------------------------------------------------
__global__ void k_buildH0(const float* __restrict__ xyz, const float* __restrict__ pts,
                          const int* __restrict__ fps, const int* __restrict__ knn,
                          int K, float* __restrict__ H) {
    int m = blockIdx.x;
    int c = threadIdx.x;
    if (c >= 68) return;
    int k  = m % K;
    int bp = m / K;
    int p  = bp % PP;
    int b  = bp / PP;
    int nb = knn[bp * KMAX + k];
    float v;
    if (c < 3) {
        int q = fps[p];
        v = xyz[(b * 3 + c) * NN + nb] - xyz[(b * 3 + c) * NN + q];
    } else if (c < 67) {
        v = pts[(b * CIN + (c - 3)) * NN + nb];
    } else {
        v = 0.0f;
    }
    H[(size_t)m * 68 + c] = v;
}

// ---------------------------------------------------------------------------
// 4) Pad weights (co x ci) -> (co x ciPad), zero-fill tail
// ---------------------------------------------------------------------------
__global__ void k_padW(const float* __restrict__ W, int co, int ci, int ciPad,
                       float* __restrict__ Wp) {
    int i = blockIdx.x * blockDim.x + threadIdx.x;
    if (i >= co * ciPad) return;
    int o = i / ciPad;
    int c = i % ciPad;
    Wp[i] = (c < ci) ? W[o * ci + c] : 0.0f;
}

// ---------------------------------------------------------------------------
// 5) GEMM via V_WMMA_F32_16X16X4_F32.
//    A: (M x CIPAD) row-major activations, B: Wp (co x CIPAD) row-major.
//    Y[m][o] = sum_c Ain[m][c] * W[o][c] + bias[o]
//    AFFINE: Ain = relu(A*scale + shift)  (previous layer BN+ReLU fused)
//    Wave tile = 64(M) x 16(O): 4 accumulators reuse one B fragment
//    per k-step (1 weight load amortized over 4 WMMAs).
//    fp32 A frag layout: lanes0-15 K={0,1}, lanes16-31 K={2,3}
// ---------------------------------------------------------------------------
template <int CIPAD, bool AFFINE>
__global__ void __launch_bounds__(128) k_gemm(const float* __restrict__ A,
                                              const float* __restrict__ Wp,
                                              const float* __restrict__ bias,
                                              const float* __restrict__ scale,
                                              const float* __restrict__ shift,
                                              float* __restrict__ Y,
                                              int M, int co) {
    int lane = threadIdx.x & 31;
    int wave = threadIdx.x >> 5;
    int m0 = blockIdx.x * 256 + wave * 64;   // 64 M-rows per wave
    int o0 = blockIdx.y * 16;
    int row  = lane & 15;     // M row (A) / O col (B)
    int half = lane >> 4;     // K half (0 -> K0,K1 ; 1 -> K2,K3)

    const float* arow = A  + (size_t)(m0 + row) * CIPAD + 2 * half;
    const float* brow = Wp + (size_t)(o0 + row) * CIPAD + 2 * half;

    v8f acc0 = {}, acc1 = {}, acc2 = {}, acc3 = {};
#pragma unroll
    for (int c0 = 0; c0 < CIPAD; c0 += 4) {
        v2f w;
        w.x = brow[c0];
        w.y = brow[c0 + 1];

        v2f a0, a1, a2, a3;
        a0.x = arow[c0];
        a0.y = arow[c0 + 1];
        a1.x = arow[c0 + 16 * CIPAD];
        a1.y = arow[c0 + 16 * CIPAD + 1];
        a2.x = arow[c0 + 32 * CIPAD];
        a2.y = arow[c0 + 32 * CIPAD + 1];
        a3.x = arow[c0 + 48 * CIPAD];
        a3.y = arow[c0 + 48 * CIPAD + 1];

        if (AFFINE) {
            int ca = c0 + 2 * half;
            float s0 = scale[ca], s1 = scale[ca + 1];
            float t0 = shift[ca], t1 = shift[ca + 1];
            a0.x = fmaxf(a0.x * s0 + t0, 0.0f);
            a0.y = fmaxf(a0.y * s1 + t1, 0.0f);
            a1.x = fmaxf(a1.x * s0 + t0, 0.0f);
            a1.y = fmaxf(a1.y * s1 + t1, 0.0f);
            a2.x = fmaxf(a2.x * s0 + t0, 0.0f);
            a2.y = fmaxf(a2.y * s1 + t1, 0.0f);
            a3.x = fmaxf(a3.x * s0 + t0, 0.0f);
            a3.y = fmaxf(a3.y * s1 + t1, 0.0f);
        }

        acc0 = __builtin_amdgcn_wmma_f32_16x16x4_f32(false, a0, false, w,
                                                     (short)0, acc0, false, false);
        acc1 = __builtin_amdgcn_wmma_f32_16x16x4_f32(false, a1, false, w,
                                                     (short)0, acc1, false, false);
        acc2 = __builtin_amdgcn_wmma_f32_16x16x4_f32(false, a2, false, w,
                                                     (short)0, acc2, false, false);
        acc3 = __builtin_amdgcn_wmma_f32_16x16x4_f32(false, a3, false, w,
                                                     (short)0, acc3, false, false);
    }

    // D layout: lanes0-15 N=lane, M=r; lanes16-31 N=lane-16, M=r+8
    int col = lane & 15;
    float bo = bias[o0 + col];
    v8f accs[4] = {acc0, acc1, acc2, acc3};
#pragma unroll
    for (int t = 0; t < 4; ++t) {
        int mb = m0 + t * 16 + 8 * half;
#pragma unroll
        for (int r = 0; r < 8; ++r) {
            Y[(size_t)(mb + r) * co + o0 + col] = accs[t][r] + bo;
        }
    }
}

// ---------------------------------------------------------------------------
// 6) BN statistics over all M rows per channel -> folded scale/shift
//    scale = g * rsqrt(var+eps); shift = be - mean*scale
// ---------------------------------------------------------------------------
__global__ void __launch_bounds__(256) k_stats(const float* __restrict__ Y, int M, int co,
                                               const float* __restrict__ g,
                                               const float* __restrict__ be,
                                               float* __restrict__ scale,
                                               float* __restrict__ shift) {
    int o = blockIdx.x;
    __shared__ float ssum[256];
    __shared__ float ssq[256];
    float s = 0.0f, q = 0.0f;
    for (int m = threadIdx.x; m < M; m += 256) {
        float v = Y[(size_t)m * co + o];
        s += v;
        q += v * v;
    }
    ssum[threadIdx.x] = s;
    ssq[threadIdx.x]  = q;
    __syncthreads();
    for (int st = 128; st > 0; st >>= 1) {
        if (threadIdx.x < st) {
            ssum[threadIdx.x] += ssum[threadIdx.x + st];
            ssq[threadIdx.x]  += ssq[threadIdx.x + st];
        }
        __syncthreads();
    }
    if (threadIdx.x == 0) {
        float mean = ssum[0] / (float)M;
        float var  = ssq[0] / (float)M - mean * mean;
        float sc   = g[o] * rsqrtf(var + EPS_BN);
        scale[o] = sc;
        shift[o] = be[o] - mean * sc;
    }
}

// ---------------------------------------------------------------------------
// 7) Final BN affine (no relu) + max over K, scatter to (B,384,P) output
// ---------------------------------------------------------------------------
__global__ void k_maxpool(const float* __restrict__ Y,
                          const float* __restrict__ scale,
                          const float* __restrict__ shift,
                          int K, int co, int chOff, float* __restrict__ out) {
    int i = blockIdx.x * blockDim.x + threadIdx.x;
    if (i >= BB * PP * co) return;
    int o  = i % co;
    int bp = i / co;
    int p  = bp % PP;
    int b  = bp / PP;
    float sc = scale[o], sh = shift[o];
    float mx = -3.4e38f;
    for (int k = 0; k < K; ++k) {
        float v = Y[((size_t)bp * K + k) * co + o];
        mx = fmaxf(mx, v * sc + sh);
    }
    out[(size_t)BB * 3 * PP + ((size_t)b * 384 + chOff + o) * PP + p] = mx;
}

// ---------------------------------------------------------------------------
// Host orchestration
// ---------------------------------------------------------------------------
extern "C" void kernel_launch(void* const* d_in, const int* in_sizes, int n_in,
                              void* d_out, int out_size, void* d_ws, size_t ws_size,
                              hipStream_t stream) {
    const float* xyz = (const float*)d_in[0];
    const float* pts = (const float*)d_in[1];
    const int*   fps = (const int*)d_in[2];
    // params flatten (JAX pytree, dict keys sorted): per layer W, b, be, g
    // index = 3 + (br*3 + l)*4 + {0:W, 1:b, 2:be, 3:g}
    float* out = (float*)d_out;

    char* ws = (char*)d_ws;
    size_t off = 0;
    int* knn = (int*)(ws + off);        off += (size_t)BB * PP * KMAX * 4;     // 2 MB
    float* Wp = (float*)(ws + off);     off += (size_t)128 * 96 * 4;           // padded weights
    float* scaleB = (float*)(ws + off); off += 128 * 4;
    float* shiftB = (float*)(ws + off); off += 128 * 4;
    off = (off + 255) & ~(size_t)255;
    float* Hbuf = (float*)(ws + off); off += (size_t)BB * PP * KMAX * 68 * 4;  // 142 MB
    float* Y0   = (float*)(ws + off); off += (size_t)BB * PP * KMAX * 128 * 4; // 268 MB
    float* Y1   = (float*)(ws + off);                                          // 268 MB

    k_newxyz<<<(BB * 3 * PP + 255) / 256, 256, 0, stream>>>(xyz, fps, out);
    k_knn<<<BB * PP, 256, 0, stream>>>(xyz, fps, knn);

    const int Ks[3] = {16, 32, 64};
    const int COs[3][3] = {{64, 64, 128}, {64, 96, 128}, {64, 96, 128}};

    for (int br = 0; br < 3; ++br) {
        int K = Ks[br];
        int M = BB * PP * K;

        k_buildH0<<<M, 128, 0, stream>>>(xyz, pts, fps, knn, K, Hbuf);

        const float* srcA = Hbuf;
        int ci = 67, ciPad = 68;
        float* Ys[2] = {Y0, Y1};

        for (int l = 0; l < 3; ++l) {
            int co = COs[br][l];
            int pi = 3 + (br * 3 + l) * 4;
            const float* W  = (const float*)d_in[pi + 0];
            const float* bb = (const float*)d_in[pi + 1];
            const float* be = (const float*)d_in[pi + 2];
            const float* g  = (const float*)d_in[pi + 3];

            k_padW<<<(co * ciPad + 255) / 256, 256, 0, stream>>>(W, co, ci, ciPad, Wp);

            float* Yout = Ys[l & 1];
            dim3 grid(M / 256, co / 16);
            // layer0: ciPad==68, no affine; layer1: ciPad==64, affine;
            // layer2: ciPad in {64,96}, affine
            if (ciPad == 68) {
                k_gemm<68, false><<<grid, 128, 0, stream>>>(srcA, Wp, bb, scaleB, shiftB,
                                                            Yout, M, co);
            } else if (ciPad == 64) {
                k_gemm<64, true><<<grid, 128, 0, stream>>>(srcA, Wp, bb, scaleB, shiftB,
                                                           Yout, M, co);
            } else { // 96
                k_gemm<96, true><<<grid, 128, 0, stream>>>(srcA, Wp, bb, scaleB, shiftB,
                                                           Yout, M, co);
            }
            k_stats<<<co, 256, 0, stream>>>(Yout, M, co, g, be, scaleB, shiftB);

            srcA = Yout;
            ci = co;
            ciPad = co;
        }

        // Final layer output is in Y0 (l=2 -> Ys[0]); stats already in scaleB/shiftB
        k_maxpool<<<(BB * PP * 128 + 255) / 256, 256, 0, stream>>>(
            Y0, scaleB, shiftB, K, 128, br * 128, out);
    }
}